// GraphStructure_33990371181042
// MI455X (gfx1250) — compile-verified
//
#include <hip/hip_runtime.h>

typedef float v2f __attribute__((ext_vector_type(2)));
typedef float v8f __attribute__((ext_vector_type(8)));
typedef unsigned int v4u __attribute__((ext_vector_type(4)));
typedef int v4i __attribute__((ext_vector_type(4)));
typedef int v8i __attribute__((ext_vector_type(8)));

#define NROWS 32768
#define DIM   128
#define KTOP  10
#define WAVES_PER_WG   8
#define ROWS_PER_WAVE  16
#define ROWS_PER_WG    (WAVES_PER_WG * ROWS_PER_WAVE)   // 128
#define NTILES         (NROWS / 16)                     // 2048
#define BSTRIDE        (DIM + 4)                        // padded row: conflict-free B reads

// ---- Tensor Data Mover: DMA one 16x128 fp32 tile (row-padded +4 DWORDs) into LDS ----
// D# per CDNA5 ISA §8.3/8.4: group0 = {count=1, lds_addr, global_addr(57b), type=2}
// group1 = {mask=0, data_size=4B, pad_enable, pad_interval=128dw, pad_amount=4dw,
//           tensor_dim0=128, tensor_dim1=32768, tile_dim0=128, tile_dim1=16, stride0=128}
__device__ static inline void tdm_load_tile(const float* gptr, unsigned lds_off) {
    unsigned long long ga = (unsigned long long)gptr;
    v4u g0;
    g0.x = 1u;                                        // count=1, user descriptor
    g0.y = lds_off;                                   // lds_addr (bytes)
    g0.z = (unsigned)(ga & 0xFFFFFFFFu);              // global_addr[31:0]
    g0.w = (unsigned)((ga >> 32) & 0x01FFFFFFu)       // global_addr[56:32]
         | (2u << 30);                                // type = 2 ("image")
    v8i g1;
    g1[0] = (int)((2u << 16)      // data_size = 4 bytes
                | (1u << 20)      // pad_enable
                | (6u << 22)      // pad_interval: 2^(6+1) = 128 DWORDs
                | (3u << 25));    // pad_amount: 3+1 = 4 DWORDs
    g1[1] = (int)(128u << 16);    // tensor_dim0 = 128 (bits 79:48, low half)
    g1[2] = (int)(0x8000u << 16); // tensor_dim0 hi=0 | tensor_dim1 = 32768 (low half)
    g1[3] = (int)(128u << 16);    // tensor_dim1 hi=0 | tile_dim0 = 128
    g1[4] = 16;                   // tile_dim1 = 16, tile_dim2 = 0
    g1[5] = 128;                  // tensor_dim0_stride = 128 elements (low 32)
    g1[6] = 0;                    // stride0 hi | tensor_dim1_stride lo
    g1[7] = 0;
    v4i z4 = {0, 0, 0, 0};
#if __has_include(<hip/amd_detail/amd_gfx1250_TDM.h>)
    v8i z8 = {0, 0, 0, 0, 0, 0, 0, 0};
    __builtin_amdgcn_tensor_load_to_lds(g0, g1, z4, z4, z8, 0);   // 6-arg toolchain
#else
    __builtin_amdgcn_tensor_load_to_lds(g0, g1, z4, z4, 0);       // 5-arg (ROCm 7.2)
#endif
}

// One workgroup = 8 wave32 = 128 rows of F. Each wave holds a 16x128 A tile in
// registers (WMMA-A layout) and streams all 2048 16-column tiles: 16x16 sims
// tile via 32 chained V_WMMA_F32_16X16X4_F32, then folds 16 candidates into
// per-row top-10 lists (lanes 0-15: most-similar, lanes 16-31: most-dissimilar).
// B tiles arrive via TDM double-buffering driven by wave 0.
__global__ __launch_bounds__(256) void bfs_topk_kernel(
    const float* __restrict__ F, int* __restrict__ out_sim, int* __restrict__ out_dis)
{
    __shared__ float Btile[2][16 * BSTRIDE];          // double-buffered DMA target
    __shared__ float Ctile[WAVES_PER_WG][16 * 17];    // per-wave C spill, padded

    const int tid  = threadIdx.x;
    const int lane = tid & 31;
    const int wave = tid >> 5;
    const int half = lane >> 4;      // 0: K-even pair / sim lanes, 1: K-odd pair / dis lanes
    const int l16  = lane & 15;

    const int rowbase = blockIdx.x * ROWS_PER_WG + wave * ROWS_PER_WAVE;
    const int myrow   = rowbase + l16;

    const unsigned ldsB0 = (unsigned)(uintptr_t)(void*)&Btile[0][0];
    const unsigned ldsB1 = (unsigned)(uintptr_t)(void*)&Btile[1][0];

    // ---- A tile (16x128) into registers, native WMMA-A layout ----
    v2f A[32];
    {
        const float* arow = F + (size_t)(rowbase + l16) * DIM + 2 * half;
        #pragma unroll
        for (int c = 0; c < 32; ++c)
            A[c] = *(const v2f*)(arow + 4 * c);
    }

    // ---- top-k register state (sorted descending; dis lanes hold -sims) ----
    float vals[KTOP];
    int   idxs[KTOP];
    #pragma unroll
    for (int k = 0; k < KTOP; ++k) { vals[k] = -3.4e38f; idxs[k] = 0; }
    const float sgn = half ? -1.0f : 1.0f;

    // ---- prime the pipeline: tile 0 -> buffer 0 ----
    if (wave == 0) tdm_load_tile(F, ldsB0);

    for (int j = 0; j < NTILES; ++j) {
        const int b = j & 1;
        if (wave == 0) {
            if (j + 1 < NTILES) {
                tdm_load_tile(F + (size_t)16 * (j + 1) * DIM, b ? ldsB0 : ldsB1);
                __builtin_amdgcn_s_wait_tensorcnt(1);   // tile j landed, j+1 in flight
            } else {
                __builtin_amdgcn_s_wait_tensorcnt(0);
            }
        }
        __syncthreads();   // buffer b valid for all waves

        // ---- 16x16 (K=128) sims tile: 32 chained v_wmma_f32_16x16x4_f32 ----
        const float* Bb = &Btile[b][0];
        v8f Cacc = {};
        #pragma unroll
        for (int c = 0; c < 32; ++c) {
            // B[K, N=l16] = F[16j+l16, K]
            v2f bv = *(const v2f*)(&Bb[l16 * BSTRIDE + 4 * c + 2 * half]);
            Cacc = __builtin_amdgcn_wmma_f32_16x16x4_f32(
                false, A[c], false, bv, (short)0, Cacc, false, false);
        }

        // ---- spill C (wave-private) so each lane can scan its row ----
        #pragma unroll
        for (int r = 0; r < 8; ++r) {
            int m = r + 8 * half;                 // row M of sims tile
            Ctile[wave][m * 17 + l16] = Cacc[r];  // N = l16
        }

        // ---- fold 16 candidates into my row's top-10 list ----
        float thresh = vals[KTOP - 1];
        for (int n = 0; n < 16; ++n) {
            float v = Ctile[wave][l16 * 17 + n];
            int   g = 16 * j + n;
            float cand = (g == myrow) ? -1.0f : sgn * v;  // self mask = -/+1.0 as in ref
            if (cand > thresh) {
                // stable sorted insert (ties keep earlier index, like jax top_k)
                #pragma unroll
                for (int k = KTOP - 1; k >= 1; --k) {
                    bool dn = cand > vals[k];
                    bool up = cand > vals[k - 1];
                    vals[k] = dn ? (up ? vals[k - 1] : cand) : vals[k];
                    idxs[k] = dn ? (up ? idxs[k - 1] : g   ) : idxs[k];
                }
                if (cand > vals[0]) { vals[0] = cand; idxs[0] = g; }
                thresh = vals[KTOP - 1];
            }
        }
        __syncthreads();   // all waves done with buffer b before TDM overwrites it
    }

    // ---- write [self, idx0..idx9] for my row ----
    int* dst = (half ? out_dis : out_sim) + (size_t)myrow * (KTOP + 1);
    dst[0] = myrow;
    #pragma unroll
    for (int k = 0; k < KTOP; ++k) dst[1 + k] = idxs[k];
}

extern "C" void kernel_launch(void* const* d_in, const int* in_sizes, int n_in,
                              void* d_out, int out_size, void* d_ws, size_t ws_size,
                              hipStream_t stream) {
    (void)in_sizes; (void)n_in; (void)d_ws; (void)ws_size; (void)out_size;
    const float* F = (const float*)d_in[0];
    // output tuple: sim table [N,11] int32, then dis table [N,11] int32
    int* out_sim = (int*)d_out;
    int* out_dis = out_sim + (size_t)NROWS * (KTOP + 1);

    dim3 grid(NROWS / ROWS_PER_WG);   // 256 workgroups
    dim3 block(32 * WAVES_PER_WG);    // 256 threads = 8 wave32
    bfs_topk_kernel<<<grid, block, 0, stream>>>(F, out_sim, out_dis);
}